// CustomLoss_62921270887106
// MI455X (gfx1250) — compile-verified
//
#include <hip/hip_runtime.h>
#include <hip/hip_bf16.h>

// CustomLoss for MI455X (gfx1250, wave32).
// Streaming reduction: 408 MB read once -> one fp32 scalar.
// Stage each 128-row chunk into LDS with CDNA5 async global->LDS b128
// copies (contiguous 512B per wave-instruction, tracked by ASYNCcnt),
// then compute one row per thread from LDS.
// Deterministic two-stage reduction, no atomics.

typedef float f4  __attribute__((ext_vector_type(4)));
typedef int   v4i __attribute__((vector_size(4 * sizeof(int))));

#define AS1 __attribute__((address_space(1)))
#define AS3 __attribute__((address_space(3)))

#if defined(__has_builtin)
#if __has_builtin(__builtin_amdgcn_global_load_async_to_lds_b128) && \
    __has_builtin(__builtin_amdgcn_s_wait_asynccnt)
#define USE_ASYNC_LDS 1
#endif
#endif

static constexpr int J     = 17;
static constexpr int ROWF  = 3 * J;          // 51 floats per row
static constexpr int ROWS  = 128;            // rows per block (== blockDim.x)
static constexpr int TILEF = ROWS * ROWF;    // 6528 floats per tile
static constexpr int TILE4 = TILEF / 4;      // 1632 float4 per tile (exact)

__device__ __forceinline__ void joint_update(float px, float py, float p,
                                             float gx, float gy, float y,
                                             float& cls, float& sq, float& cnt) {
    float lp = fmaxf(__logf(p),        -100.0f);
    float l1 = fmaxf(__logf(1.0f - p), -100.0f);
    cls += y * lp + (1.0f - y) * l1;
    float dx = px - gx;
    float dy = py - gy;
    float m  = (p >= 0.5f) ? 1.0f : 0.0f;
    sq  += m * (dx * dx + dy * dy);
    cnt += (y == 1.0f) ? 1.0f : 0.0f;
}

__global__ __launch_bounds__(ROWS) void
loss_stage1(const float* __restrict__ outp,
            const float* __restrict__ tgtp,
            int B, float2* __restrict__ partials) {
    __shared__ float lo[TILEF];
    __shared__ float lt[TILEF];

    const int tid = threadIdx.x;
    const int blk = blockIdx.x;
    const long long row0 = (long long)blk * ROWS;

    float cls = 0.0f, sq = 0.0f, cnt = 0.0f;

    if (row0 + ROWS <= B) {
        // ---- Full tile: async-stage 2 x 26112B of contiguous data into LDS.
#if defined(USE_ASYNC_LDS)
        AS1 v4i* go  = (AS1 v4i*)(outp + (size_t)row0 * ROWF);
        AS1 v4i* gt  = (AS1 v4i*)(tgtp + (size_t)row0 * ROWF);
        AS3 v4i* lo4 = (AS3 v4i*)lo;
        AS3 v4i* lt4 = (AS3 v4i*)lt;
        for (int i = tid; i < TILE4; i += ROWS) {
            __builtin_amdgcn_global_load_async_to_lds_b128(go + i, lo4 + i, 0, 0);
            __builtin_amdgcn_global_load_async_to_lds_b128(gt + i, lt4 + i, 0, 0);
        }
        __builtin_amdgcn_s_wait_asynccnt(0);
        __syncthreads();
#else
        // Fallback staging: NT b128 loads through VGPRs into LDS.
        const f4* go = (const f4*)(outp + (size_t)row0 * ROWF);
        const f4* gt = (const f4*)(tgtp + (size_t)row0 * ROWF);
        f4* lo4 = (f4*)lo;
        f4* lt4 = (f4*)lt;
        for (int i = tid; i < TILE4; i += ROWS) {
            lo4[i] = __builtin_nontemporal_load(go + i);
            lt4[i] = __builtin_nontemporal_load(gt + i);
        }
        __syncthreads();
#endif
        // ---- Compute: one row per thread from LDS (stride 51 dwords is odd
        // -> conflict-free b32 reads across a wave32).
        const float* o = lo + tid * ROWF;
        const float* t = lt + tid * ROWF;
#pragma unroll
        for (int j = 0; j < J; ++j)
            joint_update(o[3 * j], o[3 * j + 1], o[3 * j + 2],
                         t[3 * j], t[3 * j + 1], t[3 * j + 2],
                         cls, sq, cnt);
    } else if (row0 + tid < B) {
        // ---- Tail tile: direct global reads, one row per thread.
        const float* o = outp + (size_t)(row0 + tid) * ROWF;
        const float* t = tgtp + (size_t)(row0 + tid) * ROWF;
#pragma unroll
        for (int j = 0; j < J; ++j)
            joint_update(o[3 * j], o[3 * j + 1], o[3 * j + 2],
                         t[3 * j], t[3 * j + 1], t[3 * j + 2],
                         cls, sq, cnt);
    }

    float reg = 0.5f * sq / (1.0f + cnt);
    if (row0 + tid >= B) { reg = 0.0f; cls = 0.0f; }   // inactive rows

    // ---- Deterministic block reduction (fixed tree, no atomics).
    __shared__ float sC[ROWS];
    __shared__ float sR[ROWS];
    sC[tid] = cls;
    sR[tid] = reg;
    __syncthreads();
#pragma unroll
    for (int s = ROWS / 2; s > 0; s >>= 1) {
        if (tid < s) {
            sC[tid] += sC[tid + s];
            sR[tid] += sR[tid + s];
        }
        __syncthreads();
    }
    if (tid == 0) {
        float2 p;
        p.x = sC[0];
        p.y = sR[0];
        partials[blk] = p;
    }
}

__global__ __launch_bounds__(256) void
loss_stage2(const float2* __restrict__ partials, int nparts,
            float* __restrict__ out, int B) {
    const int tid = threadIdx.x;
    float c = 0.0f, r = 0.0f;
    for (int i = tid; i < nparts; i += 256) {     // fixed order per thread
        float2 p = partials[i];
        c += p.x;
        r += p.y;
    }
    __shared__ float sC[256];
    __shared__ float sR[256];
    sC[tid] = c;
    sR[tid] = r;
    __syncthreads();
#pragma unroll
    for (int s = 128; s > 0; s >>= 1) {
        if (tid < s) {
            sC[tid] += sC[tid + s];
            sR[tid] += sR[tid + s];
        }
        __syncthreads();
    }
    if (tid == 0) {
        const float Bf = (float)B;
        float loss_class = -sC[0] / (Bf * (float)J);
        float loss_reg   =  sR[0] / Bf;
        out[0] = loss_class + loss_reg;
    }
}

extern "C" void kernel_launch(void* const* d_in, const int* in_sizes, int n_in,
                              void* d_out, int out_size, void* d_ws, size_t ws_size,
                              hipStream_t stream) {
    const float* outp = (const float*)d_in[0];   // (B, 17, 3) fp32
    const float* tgtp = (const float*)d_in[1];   // (B, 17, 3) fp32
    const int B       = in_sizes[0] / ROWF;

    const int nblocks = (B + ROWS - 1) / ROWS;   // one 128-row tile per block

    float2* partials = (float2*)d_ws;            // nblocks * 8 bytes

    loss_stage1<<<nblocks, ROWS, 0, stream>>>(outp, tgtp, B, partials);
    loss_stage2<<<1, 256, 0, stream>>>(partials, nblocks, (float*)d_out, B);
}